// Attention_13451837571855
// MI455X (gfx1250) — compile-verified
//
#include <hip/hip_runtime.h>
#include <hip/hip_bf16.h>

typedef __attribute__((ext_vector_type(2))) float v2f;
typedef __attribute__((ext_vector_type(4))) float v4f;
typedef __attribute__((ext_vector_type(8))) float v8f;

#define NB   16      // batch
#define NK   4096    // keys
#define ED   1024    // embed dim
#define AD   256     // attn dim
#define NCHUNK 32    // j-chunks for attn@v partial reduction (4096/128)

// ---------------------------------------------------------------------------
// Generic 16xN fp32 WMMA GEMM:  C[16,N] = A[16,K] @ B + bias
// B element (k,n) addressed as Bm[k*sbk + n*sbn]  (handles normal + transpose)
// One wave computes one 16-wide N tile using V_WMMA_F32_16X16X4_F32, K step 4.
// Launch with blockDim=128 (4 waves), grid.x * 4 == N/16 tiles exactly.
// ---------------------------------------------------------------------------
__global__ __launch_bounds__(128) void gemm16_wmma_kernel(
    const float* __restrict__ A, int lda,
    const float* __restrict__ Bm, int sbk, int sbn,
    const float* __restrict__ bias,
    float* __restrict__ C, int ldc, int K) {
  const int wave = threadIdx.x >> 5;
  const int lane = threadIdx.x & 31;
  const int tile = blockIdx.x * (blockDim.x >> 5) + wave;
  const int n0   = tile * 16;
  const int hi   = lane >> 4;      // lane half selects K pair (ISA 16x4 f32 A layout)
  const int m    = lane & 15;      // A row / C column-lane
  const int n    = n0 + m;         // B/C column for this lane

  v8f acc = {0.f, 0.f, 0.f, 0.f, 0.f, 0.f, 0.f, 0.f};
  for (int k0 = 0; k0 < K; k0 += 4) {
    const int ka = k0 + 2 * hi;    // VGPR0: K=ka, VGPR1: K=ka+1
    v2f a = *((const v2f*)(A + (size_t)m * lda + ka));
    v2f b;
    b.x = Bm[(size_t)ka       * sbk + (size_t)n * sbn];
    b.y = Bm[(size_t)(ka + 1) * sbk + (size_t)n * sbn];
    acc = __builtin_amdgcn_wmma_f32_16x16x4_f32(
        /*neg_a=*/false, a, /*neg_b=*/false, b,
        /*c_mod=*/(short)0, acc, /*reuse_a=*/false, /*reuse_b=*/false);
  }
  const float bb = bias ? bias[n] : 0.0f;
#pragma unroll
  for (int i = 0; i < 8; ++i) {
    const int r = i + 8 * hi;      // C/D layout: VGPR i -> rows i (lo half), i+8 (hi half)
    C[(size_t)r * ldc + n] = acc[i] + bb;
  }
}

// ---------------------------------------------------------------------------
// scores[b,j] = (qt[b] . k[b,j]) / 16   -- streams k (268 MB) with NT loads.
// Block: 256 threads = 8 waves; wave handles 8 consecutive j rows; each row is
// a fully coalesced 4 KB read (8 iters x 32 lanes x float4).
// Grid: dim3(64, 16)
// ---------------------------------------------------------------------------
__global__ __launch_bounds__(256) void scores_kernel(
    const float* __restrict__ kmat, const float* __restrict__ qt,
    float* __restrict__ scores) {
  __shared__ float qs[ED];
  const int b = blockIdx.y;
  const int jbase = blockIdx.x * 64;
  for (int i = threadIdx.x; i < ED; i += 256) qs[i] = qt[b * ED + i];
  __syncthreads();
  const int wave = threadIdx.x >> 5;
  const int lane = threadIdx.x & 31;
  for (int jj = 0; jj < 8; ++jj) {
    const int j = jbase + wave * 8 + jj;
    const float* row = kmat + ((size_t)b * NK + j) * ED;
    float acc = 0.f;
#pragma unroll
    for (int it = 0; it < 8; ++it) {
      const int d = it * 128 + lane * 4;
      v4f kv = __builtin_nontemporal_load((const v4f*)(row + d));
      acc += kv.x * qs[d] + kv.y * qs[d + 1] + kv.z * qs[d + 2] + kv.w * qs[d + 3];
    }
#pragma unroll
    for (int off = 16; off > 0; off >>= 1) acc += __shfl_xor(acc, off, 32);
    if (lane == 0) scores[(size_t)b * NK + j] = acc * 0.0625f;  // 1/sqrt(256)
  }
}

// ---------------------------------------------------------------------------
// Softmax over NK per batch. One block per batch, 256 threads x 16 values.
// ---------------------------------------------------------------------------
__global__ __launch_bounds__(256) void softmax_kernel(
    const float* __restrict__ scores, float* __restrict__ attn) {
  __shared__ float red[256];
  const int b = blockIdx.x;
  const int t = threadIdx.x;
  float vals[16];
  float m = -3.4e38f;
#pragma unroll
  for (int i = 0; i < 16; ++i) {
    vals[i] = scores[(size_t)b * NK + i * 256 + t];
    m = fmaxf(m, vals[i]);
  }
  red[t] = m;
  __syncthreads();
  for (int s = 128; s > 0; s >>= 1) {
    if (t < s) red[t] = fmaxf(red[t], red[t + s]);
    __syncthreads();
  }
  const float gmax = red[0];
  __syncthreads();
  float sum = 0.f;
#pragma unroll
  for (int i = 0; i < 16; ++i) {
    vals[i] = __expf(vals[i] - gmax);
    sum += vals[i];
  }
  red[t] = sum;
  __syncthreads();
  for (int s = 128; s > 0; s >>= 1) {
    if (t < s) red[t] += red[t + s];
    __syncthreads();
  }
  const float inv = 1.0f / red[0];
#pragma unroll
  for (int i = 0; i < 16; ++i)
    attn[(size_t)b * NK + i * 256 + t] = vals[i] * inv;
}

// ---------------------------------------------------------------------------
// partials[b,c,:] = sum_{j in chunk c} attn[b,j] * v[b,j,:]  -- streams v.
// Block: 256 threads x float4 covers all 1024 e; per-j read = coalesced 4 KB.
// Grid: dim3(NCHUNK, 16); chunk = 128 j.
// ---------------------------------------------------------------------------
__global__ __launch_bounds__(256) void av_partial_kernel(
    const float* __restrict__ vmat, const float* __restrict__ attn,
    float* __restrict__ partials) {
  __shared__ float aw[128];
  const int b = blockIdx.y;
  const int c = blockIdx.x;
  const int t = threadIdx.x;
  const int jbase = c * 128;
  if (t < 128) aw[t] = attn[(size_t)b * NK + jbase + t];
  __syncthreads();
  const int e = t * 4;
  v4f acc = {0.f, 0.f, 0.f, 0.f};
  const float* base = vmat + ((size_t)b * NK + jbase) * ED + e;
  for (int jj = 0; jj < 128; ++jj) {
    v4f vv = __builtin_nontemporal_load((const v4f*)(base + (size_t)jj * ED));
    const float w = aw[jj];
    acc.x += w * vv.x; acc.y += w * vv.y; acc.z += w * vv.z; acc.w += w * vv.w;
  }
  *((v4f*)(partials + ((size_t)(b * NCHUNK + c)) * ED + e)) = acc;
}

// u[b,e] = sum_c partials[b,c,e].  Grid: dim3(4,16), block 256.
__global__ __launch_bounds__(256) void reduce_u_kernel(
    const float* __restrict__ partials, float* __restrict__ u) {
  const int b = blockIdx.y;
  const int e = blockIdx.x * 256 + threadIdx.x;
  float s = 0.f;
#pragma unroll
  for (int c = 0; c < NCHUNK; ++c)
    s += partials[((size_t)(b * NCHUNK + c)) * ED + e];
  u[(size_t)b * ED + e] = s;
}

// ---------------------------------------------------------------------------
extern "C" void kernel_launch(void* const* d_in, const int* in_sizes, int n_in,
                              void* d_out, int out_size, void* d_ws, size_t ws_size,
                              hipStream_t stream) {
  const float* q    = (const float*)d_in[0];  // [16,1,1024]
  const float* k    = (const float*)d_in[1];  // [16,4096,1024]
  const float* v    = (const float*)d_in[2];  // [16,4096,1024]
  const float* W_kq = (const float*)d_in[3];  // [1024,256]
  const float* b_kq = (const float*)d_in[4];  // [256]
  const float* W_v  = (const float*)d_in[5];  // [1024,1024]
  const float* b_v  = (const float*)d_in[6];  // [1024]
  float* out = (float*)d_out;                 // [16,1024]

  float* qp       = (float*)d_ws;             // 16*256
  float* qt       = qp + NB * AD;             // 16*1024
  float* scores   = qt + NB * ED;             // 16*4096
  float* attn     = scores + NB * NK;         // 16*4096
  float* partials = attn + NB * NK;           // 16*32*1024
  float* u        = partials + (size_t)NB * NCHUNK * ED;  // 16*1024

  // qp[16,256] = q @ W_kq + b_kq          (16 tiles -> 4 blocks x 4 waves)
  gemm16_wmma_kernel<<<dim3(4), 128, 0, stream>>>(
      q, ED, W_kq, AD, 1, b_kq, qp, AD, ED);
  // qt[16,1024] = qp @ W_kq^T             (64 tiles -> 16 blocks)
  gemm16_wmma_kernel<<<dim3(16), 128, 0, stream>>>(
      qp, AD, W_kq, 1, AD, nullptr, qt, ED, AD);
  // scores[16,4096] = (qt . k) / 16       (streams k, NT)
  scores_kernel<<<dim3(64, NB), 256, 0, stream>>>(k, qt, scores);
  // attn = softmax(scores)
  softmax_kernel<<<NB, 256, 0, stream>>>(scores, attn);
  // u[16,1024] = attn @ v                 (streams v, NT; two-stage reduction)
  av_partial_kernel<<<dim3(NCHUNK, NB), 256, 0, stream>>>(v, attn, partials);
  reduce_u_kernel<<<dim3(4, NB), 256, 0, stream>>>(partials, u);
  // out[16,1024] = u @ W_v + b_v          (64 tiles -> 16 blocks)
  gemm16_wmma_kernel<<<dim3(16), 128, 0, stream>>>(
      u, ED, W_v, ED, 1, b_v, out, ED, ED);
}